// GraphSAGE_16638703305427
// MI455X (gfx1250) — compile-verified
//
#include <hip/hip_runtime.h>

// ---------------- problem constants (from reference) ----------------
#define N0V 200000
#define N1V 50000
#define N2V 12500
#define N3V 3200
#define E0V 500000
#define E1V 125000
#define E2V 32000
#define DIN 128
#define DH  256
#define DOUT 128

typedef float v2f __attribute__((ext_vector_type(2)));
typedef float v8f __attribute__((ext_vector_type(8)));

// ---------------- utility kernels ----------------
__global__ void zero_f32(float* __restrict__ p, int n) {
  int i = blockIdx.x * blockDim.x + threadIdx.x;
  if (i < n) p[i] = 0.0f;
}

__global__ void count_edges(const int* __restrict__ dst, float* __restrict__ cnt, int E) {
  int e = blockIdx.x * blockDim.x + threadIdx.x;
  if (e < E) atomicAdd(&cnt[dst[e]], 1.0f);
}

__global__ void inv_count(float* __restrict__ cnt, int n) {
  int i = blockIdx.x * blockDim.x + threadIdx.x;
  if (i < n) cnt[i] = 1.0f / fmaxf(cnt[i], 1.0f);
}

// Scatter-add of messages: s[dst[e], :] += x[src[e], :]
// One thread per (edge, 4-feature chunk); float4 gather (L2-resident tables),
// 4x global_atomic_add_f32 scatter.
template<int D>
__global__ void scatter_add(const float* __restrict__ x, const int* __restrict__ src,
                            const int* __restrict__ dst, float* __restrict__ s, int E) {
  const int CPE = D / 4;
  int idx = blockIdx.x * blockDim.x + threadIdx.x;
  if (idx >= E * CPE) return;
  int e = idx / CPE;
  int c = (idx - e * CPE) * 4;
  int sr = src[e];
  int dr = dst[e];
  const float4 v = *(const float4*)(x + (size_t)sr * D + c);
  float* o = s + (size_t)dr * D + c;
  atomicAdd(o + 0, v.x);
  atomicAdd(o + 1, v.y);
  atomicAdd(o + 2, v.z);
  atomicAdd(o + 3, v.w);
}

// ---------------- fused SAGE GEMM ----------------
// out[n, DO] = relu?( (s .* invc) @ wl.T  +  xt @ wr.T  + bias )
// One block = one 16-row panel. blockDim = (32, DO/16): wave y owns column
// tile [16*y, 16*y+16). A-panel [mean | x_tgt] (16 x 2*DI) staged in LDS with
// +2 float padding per row (bank rotation -> conflict-free ds_load_b64).
// K is consumed 4 at a time with V_WMMA_F32_16X16X4_F32 (full fp32 math).
template<int DI, int DO, bool RELU>
__global__ __launch_bounds__(32 * (DO / 16))
void sage_gemm(const float* __restrict__ s,     // [n, DI] segment sums
               const float* __restrict__ invc,  // [n]    1/max(cnt,1)
               const float* __restrict__ xt,    // [n, DI] target features
               const float* __restrict__ wl,    // [DO, DI]
               const float* __restrict__ wr,    // [DO, DI]
               const float* __restrict__ bias,  // [DO]
               float* __restrict__ out,         // [n, DO]
               int n) {
  constexpr int K2  = 2 * DI;
  constexpr int STR = K2 + 2;              // padded LDS row stride (floats)
  __shared__ float lA[16 * STR];

  const int lane = threadIdx.x;            // wave32
  const int wv   = threadIdx.y;
  const int row0 = blockIdx.x * 16;
  const int col0 = wv * 16;
  const int tid  = wv * 32 + lane;
  const int T    = (DO / 16) * 32;

  // ---- stage A = [mean | x_tgt] for 16 rows into LDS ----
  for (int i = tid; i < 16 * K2; i += T) {
    int r = i / K2;
    int c = i - r * K2;
    int grow = row0 + r;
    float v = 0.0f;
    if (grow < n) {
      if (c < DI) v = s[(size_t)grow * DI + c] * invc[grow];
      else        v = xt[(size_t)grow * DI + (c - DI)];
    }
    lA[r * STR + c] = v;
  }
  __syncthreads();

  // ---- WMMA fragment addressing (ISA 16x16x4 f32 layout) ----
  // A: lanes 0-15 -> M=lane, K pair {k,k+1}; lanes 16-31 -> M=lane-16, {k+2,k+3}
  // B: lanes 0-15 -> N=lane, K pair {k,k+1}; lanes 16-31 -> N=lane-16, {k+2,k+3}
  const int m    = lane & 15;
  const int koff = (lane < 16) ? 0 : 2;
  const int col  = col0 + m;
  const float* __restrict__ pl   = wl + (size_t)col * DI;  // wl row = W column
  const float* __restrict__ pr   = wr + (size_t)col * DI;
  const float* __restrict__ arow = lA + m * STR + koff;

  const float bb = bias[col];
  v8f acc = {bb, bb, bb, bb, bb, bb, bb, bb};

  // K-half 1: mean @ wl.T
#pragma unroll 4
  for (int k = 0; k < DI; k += 4) {
    v2f a = *(const v2f*)(arow + k);
    v2f b = *(const v2f*)(pl + k + koff);
    acc = __builtin_amdgcn_wmma_f32_16x16x4_f32(
        /*neg_a=*/false, a, /*neg_b=*/false, b,
        /*c_mod=*/(short)0, acc, /*reuse_a=*/false, /*reuse_b=*/false);
  }
  // K-half 2: x_tgt @ wr.T
#pragma unroll 4
  for (int k = 0; k < DI; k += 4) {
    v2f a = *(const v2f*)(arow + DI + k);
    v2f b = *(const v2f*)(pr + k + koff);
    acc = __builtin_amdgcn_wmma_f32_16x16x4_f32(
        false, a, false, b, (short)0, acc, false, false);
  }

  // ---- epilogue: C/D layout -> row = row0 + v + (lane<16 ? 0 : 8) ----
  const int rbase = row0 + ((lane < 16) ? 0 : 8);
#pragma unroll
  for (int v = 0; v < 8; ++v) {
    int row = rbase + v;
    if (row < n) {
      float o = acc[v];
      if (RELU) o = fmaxf(o, 0.0f);
      out[(size_t)row * DO + col] = o;
    }
  }
}

// ---------------- host-side orchestration ----------------
static inline int cdiv(int a, int b) { return (a + b - 1) / b; }

extern "C" void kernel_launch(void* const* d_in, const int* in_sizes, int n_in,
                              void* d_out, int out_size, void* d_ws, size_t ws_size,
                              hipStream_t stream) {
  (void)in_sizes; (void)n_in; (void)out_size; (void)ws_size;

  const float* x    = (const float*)d_in[0];
  const int*   src0 = (const int*)d_in[1];
  const int*   dst0 = (const int*)d_in[2];
  const int*   src1 = (const int*)d_in[3];
  const int*   dst1 = (const int*)d_in[4];
  const int*   src2 = (const int*)d_in[5];
  const int*   dst2 = (const int*)d_in[6];
  const float* wl0  = (const float*)d_in[7];
  const float* wr0  = (const float*)d_in[8];
  const float* b0   = (const float*)d_in[9];
  const float* wl1  = (const float*)d_in[10];
  const float* wr1  = (const float*)d_in[11];
  const float* b1   = (const float*)d_in[12];
  const float* wl2  = (const float*)d_in[13];
  const float* wr2  = (const float*)d_in[14];
  const float* b2   = (const float*)d_in[15];
  float* out = (float*)d_out;

  // workspace carve-up (floats), 256-elem aligned
  float* ws = (float*)d_ws;
  size_t off = 0;
  auto take = [&](size_t nelem) {
    float* p = ws + off;
    off += (nelem + 255) & ~(size_t)255;
    return p;
  };
  float* s0 = take((size_t)N1V * DIN);
  float* c0 = take(N1V);
  float* h0 = take((size_t)N1V * DH);
  float* s1 = take((size_t)N2V * DH);
  float* c1 = take(N2V);
  float* h1 = take((size_t)N2V * DH);
  float* s2 = take((size_t)N3V * DH);
  float* c2 = take(N3V);

  const int TB = 256;

  // ---------- layer 0: x(N0,128) --E0--> h0(N1,256), relu ----------
  zero_f32<<<cdiv(N1V * DIN, TB), TB, 0, stream>>>(s0, N1V * DIN);
  zero_f32<<<cdiv(N1V, TB), TB, 0, stream>>>(c0, N1V);
  count_edges<<<cdiv(E0V, TB), TB, 0, stream>>>(dst0, c0, E0V);
  scatter_add<DIN><<<cdiv(E0V * (DIN / 4), TB), TB, 0, stream>>>(x, src0, dst0, s0, E0V);
  inv_count<<<cdiv(N1V, TB), TB, 0, stream>>>(c0, N1V);
  sage_gemm<DIN, DH, true><<<cdiv(N1V, 16), dim3(32, DH / 16), 0, stream>>>(
      s0, c0, x, wl0, wr0, b0, h0, N1V);

  // ---------- layer 1: h0(N1,256) --E1--> h1(N2,256), relu ----------
  zero_f32<<<cdiv(N2V * DH, TB), TB, 0, stream>>>(s1, N2V * DH);
  zero_f32<<<cdiv(N2V, TB), TB, 0, stream>>>(c1, N2V);
  count_edges<<<cdiv(E1V, TB), TB, 0, stream>>>(dst1, c1, E1V);
  scatter_add<DH><<<cdiv(E1V * (DH / 4), TB), TB, 0, stream>>>(h0, src1, dst1, s1, E1V);
  inv_count<<<cdiv(N2V, TB), TB, 0, stream>>>(c1, N2V);
  sage_gemm<DH, DH, true><<<cdiv(N2V, 16), dim3(32, DH / 16), 0, stream>>>(
      s1, c1, h0, wl1, wr1, b1, h1, N2V);

  // ---------- layer 2: h1(N2,256) --E2--> out(N3,128), no relu ----------
  zero_f32<<<cdiv(N3V * DH, TB), TB, 0, stream>>>(s2, N3V * DH);
  zero_f32<<<cdiv(N3V, TB), TB, 0, stream>>>(c2, N3V);
  count_edges<<<cdiv(E2V, TB), TB, 0, stream>>>(dst2, c2, E2V);
  scatter_add<DH><<<cdiv(E2V * (DH / 4), TB), TB, 0, stream>>>(h1, src2, dst2, s2, E2V);
  inv_count<<<cdiv(N3V, TB), TB, 0, stream>>>(c2, N3V);
  sage_gemm<DH, DOUT, false><<<cdiv(N3V, 16), dim3(32, DOUT / 16), 0, stream>>>(
      s2, c2, h1, wl2, wr2, b2, out, N3V);
}